// ETB_69346541961683
// MI455X (gfx1250) — compile-verified
//
#include <hip/hip_runtime.h>
#include <math.h>

// ---------------------------------------------------------------------------
// ETB block for MI455X (gfx1250, wave32). All matmul-shaped work (DFT-as-GEMM
// FFTs, attention einsums, 1x1 convs) runs through a single batched complex
// GEMM kernel built on V_WMMA_F32_16X16X4_F32, with explicit global
// address-space accesses (global_load/global_store, no flat/DScnt coupling).
// ---------------------------------------------------------------------------

typedef float v2f __attribute__((ext_vector_type(2)));
typedef float v8f __attribute__((ext_vector_type(8)));

// explicit global address space pointers -> force global_load_* / global_store_*
typedef const float __attribute__((address_space(1)))* gcf;
typedef float __attribute__((address_space(1)))*       gf;

#define PI_F 3.14159265358979323846f
#define BNS  0.9999950000374997f   // 1/sqrt(1+1e-5)

static constexpr int  Bn = 8, Cn = 128, Hd = 4;
static constexpr int  Nn = 6400;             // 80*80
static constexpr long PLANE = (long)Cn * Nn; // per-image (128,6400)
static constexpr long TOT   = (long)Bn * PLANE;
static constexpr long CHROW = 32L * Nn;      // per-(b,head) [32,6400] stride

// ---------------------------------------------------------------------------
// Batched (complex) GEMM. Each wave computes a 16 x (16*NT) C tile; K stepped
// by 4 via V_WMMA_F32_16X16X4_F32. The A fragment (and its negation for the
// complex MAC) is loaded once per K-step and reused across the NT B-subtiles.
// AIM/BIM: A/B has imaginary plane. TRB: B accessed transposed.
// BOUND=false: all dims assumed tile-aligned -> zero guards, branch-free loop.
// BOUND=true : clamp+mask (multiply by 0/1) instead of exec-mask branches.
// ---------------------------------------------------------------------------
template<bool AIM, bool BIM, bool TRB, int NT, bool BOUND>
__global__ __launch_bounds__(32)
void cgemm16(const float* __restrict__ Are, const float* __restrict__ Aim,
             const float* __restrict__ Bre, const float* __restrict__ Bim,
             float* __restrict__ Cre, float* __restrict__ Cim,
             int M, int N, int K,
             long sAb, int lda, long sBb, int ldb, long sCb, int ldc,
             const float* __restrict__ bias, int accumulate)
{
    const int lane = (int)(threadIdx.x & 31u);
    const int hw   = lane >> 4;      // half-wave select
    const int l15  = lane & 15;
    const int n0   = blockIdx.x * (16 * NT);
    const int m0   = blockIdx.y * 16;
    const long g   = blockIdx.z;

    int   rowA = m0 + l15;
    float am   = 1.0f;
    if (BOUND && rowA >= M) { rowA = M - 1; am = 0.0f; }

    gcf pAr = (gcf)(Are + g * sAb + (long)rowA * lda + hw * 2);
    gcf pAi = AIM ? (gcf)(Aim + g * sAb + (long)rowA * lda + hw * 2) : (gcf)0;

    gcf pBr[NT];
    gcf pBi[NT];
    float bm[NT];
#pragma unroll
    for (int t = 0; t < NT; ++t) {
        int colB = n0 + 16 * t + l15;
        bm[t] = 1.0f;
        if (BOUND && colB >= N) { colB = N - 1; bm[t] = 0.0f; }
        if (TRB) {
            pBr[t] = (gcf)(Bre + g * sBb + (long)colB * ldb + hw * 2);
            if (BIM) pBi[t] = (gcf)(Bim + g * sBb + (long)colB * ldb + hw * 2);
        } else {
            pBr[t] = (gcf)(Bre + g * sBb + (long)(hw * 2) * ldb + colB);
            if (BIM) pBi[t] = (gcf)(Bim + g * sBb + (long)(hw * 2) * ldb + colB);
        }
    }

    v8f cr[NT] = {};
    v8f ci[NT] = {};

    for (int k0 = 0; k0 < K; k0 += 4) {
        v2f ar, ai, nai;
        ar[0] = pAr[0]; ar[1] = pAr[1]; pAr += 4;
        if (BOUND) { ar[0] *= am; ar[1] *= am; }
        if (AIM) {
            ai[0] = pAi[0]; ai[1] = pAi[1]; pAi += 4;
            if (BOUND) { ai[0] *= am; ai[1] *= am; }
            nai[0] = -ai[0]; nai[1] = -ai[1];
        }
#pragma unroll
        for (int t = 0; t < NT; ++t) {
            v2f br, bi;
            if (TRB) {
                br[0] = pBr[t][0]; br[1] = pBr[t][1]; pBr[t] += 4;
                if (BIM) { bi[0] = pBi[t][0]; bi[1] = pBi[t][1]; pBi[t] += 4; }
            } else {
                br[0] = pBr[t][0]; br[1] = pBr[t][ldb]; pBr[t] += (long)4 * ldb;
                if (BIM) { bi[0] = pBi[t][0]; bi[1] = pBi[t][ldb]; pBi[t] += (long)4 * ldb; }
            }
            if (BOUND) {
                br[0] *= bm[t]; br[1] *= bm[t];
                if (BIM) { bi[0] *= bm[t]; bi[1] *= bm[t]; }
            }
            cr[t] = __builtin_amdgcn_wmma_f32_16x16x4_f32(false, ar, false, br, (short)0, cr[t], false, false);
            if (AIM && BIM) {
                cr[t] = __builtin_amdgcn_wmma_f32_16x16x4_f32(false, nai, false, bi, (short)0, cr[t], false, false);
                ci[t] = __builtin_amdgcn_wmma_f32_16x16x4_f32(false, ar,  false, bi, (short)0, ci[t], false, false);
                ci[t] = __builtin_amdgcn_wmma_f32_16x16x4_f32(false, ai,  false, br, (short)0, ci[t], false, false);
            } else if (AIM) {
                ci[t] = __builtin_amdgcn_wmma_f32_16x16x4_f32(false, ai, false, br, (short)0, ci[t], false, false);
            } else if (BIM) {
                ci[t] = __builtin_amdgcn_wmma_f32_16x16x4_f32(false, ar, false, bi, (short)0, ci[t], false, false);
            }
        }
    }

    gf outR = (gf)Cre;
    gf outI = (gf)Cim;
#pragma unroll
    for (int t = 0; t < NT; ++t) {
#pragma unroll
        for (int i = 0; i < 8; ++i) {
            const int m = m0 + i + hw * 8;
            const int n = n0 + 16 * t + l15;
            if (!BOUND || (m < M && n < N)) {
                const long idx = g * sCb + (long)m * ldc + n;
                float vr = cr[t][i];
                if (bias)       vr += bias[m];
                if (accumulate) vr += outR[idx];
                outR[idx] = vr;
                if (AIM || BIM) {
                    float vi = ci[t][i];
                    if (accumulate) vi += outI[idx];
                    outI[idx] = vi;
                }
            }
        }
    }
}

// ---------------------------------------------------------------------------
// DFT / twiddle tables (computed on-device, deterministic).
// F80 = exp(-2pi i jk/80) ; iF80 = exp(+2pi i jk/80) ; iF32 = exp(+2pi i jk/32)
// TW[k1][n2] = exp(+2pi i k1 n2 / 6400) (inverse 6400-pt Cooley-Tukey twiddle)
// ---------------------------------------------------------------------------
__global__ void etb_tables(float* F80r, float* F80i, float* iF80r, float* iF80i,
                           float* TWr, float* TWi, float* iF32r, float* iF32i)
{
    int t = blockIdx.x * 256 + threadIdx.x;
    if (t < 6400) {
        int j = t / 80, k = t % 80;
        int m = (j * k) % 80;
        float a = -2.0f * PI_F * (float)m / 80.0f;
        F80r[t]  = cosf(a);  F80i[t]  = sinf(a);
        iF80r[t] = cosf(-a); iF80i[t] = sinf(-a);
        int mm = (j * k) % 6400;
        float b = 2.0f * PI_F * (float)mm / 6400.0f;
        TWr[t] = cosf(b); TWi[t] = sinf(b);
    }
    if (t < 1024) {
        int j = t / 32, k = t % 32;
        int m = (j * k) % 32;
        float a = 2.0f * PI_F * (float)m / 32.0f;
        iF32r[t] = cosf(a); iF32i[t] = sinf(a);
    }
}

// Restormer WithBias LayerNorm over channel dim (128), per pixel.
__global__ void etb_ln(const float* __restrict__ x, const float* __restrict__ w,
                       const float* __restrict__ b, float* __restrict__ y, long total)
{
    long p = (long)blockIdx.x * blockDim.x + threadIdx.x;
    if (p >= total) return;
    long bb = p / Nn; int n = (int)(p % Nn);
    const float* px = x + bb * PLANE + n;
    float s = 0.f, s2 = 0.f;
    for (int c = 0; c < Cn; ++c) { float v = px[(long)c * Nn]; s += v; s2 += v * v; }
    float mu  = s / Cn;
    float var = s2 / Cn - mu * mu;
    float inv = rsqrtf(var + 1e-5f);
    float* py = y + bb * PLANE + n;
    for (int c = 0; c < Cn; ++c)
        py[(long)c * Nn] = (px[(long)c * Nn] - mu) * inv * w[c] + b[c];
}

// L2 normalize rows of length n (optionally complex). In-place safe.
__global__ void etb_l2norm(const float* __restrict__ sr, const float* __restrict__ si,
                           float* __restrict__ dr, float* __restrict__ di, int n)
{
    __shared__ float sm[256];
    long base = (long)blockIdx.x * n;
    float s = 0.f;
    for (int i = threadIdx.x; i < n; i += 256) {
        float r = sr[base + i]; s += r * r;
        if (si) { float q = si[base + i]; s += q * q; }
    }
    sm[threadIdx.x] = s; __syncthreads();
    for (int o = 128; o > 0; o >>= 1) {
        if ((int)threadIdx.x < o) sm[threadIdx.x] += sm[threadIdx.x + o];
        __syncthreads();
    }
    float inv = 1.0f / fmaxf(sqrtf(sm[0]), 1e-12f);
    for (int i = threadIdx.x; i < n; i += 256) {
        dr[base + i] = sr[base + i] * inv;
        if (si) di[base + i] = si[base + i] * inv;
    }
}

// Row softmax over 32 entries, one wave per row; pre-scale by per-head temp.
__global__ void etb_softmax32(float* __restrict__ a, const float* __restrict__ temp, int rows)
{
    int row = blockIdx.x;
    if (row >= rows) return;
    int lane = (int)(threadIdx.x & 31u);
    long base = (long)row * 32;
    float t = temp[(row >> 5) % Hd];
    float v = a[base + lane] * t;
    float m = v;
    for (int o = 16; o > 0; o >>= 1) m = fmaxf(m, __shfl_xor(m, o, 32));
    float e = expf(v - m);
    float s = e;
    for (int o = 16; o > 0; o >>= 1) s += __shfl_xor(s, o, 32);
    a[base + lane] = e / s;
}

// Grouped KxK conv, pad same-style; one thread per output element.
__global__ void etb_gconv(const float* __restrict__ in, const float* __restrict__ wgt,
                          float* __restrict__ out, int Cin, int Cout, int icpg, int ocpg,
                          int ksz, int pad, int ocBase, int outC, long total)
{
    long idx = (long)blockIdx.x * blockDim.x + threadIdx.x;
    if (idx >= total) return;
    int n = (int)(idx % Nn); long r = idx / Nn;
    int oc = (int)(r % Cout); int b = (int)(r / Cout);
    int y = n / 80, x0 = n % 80;
    int g = oc / ocpg;
    const float* wp = wgt + (long)oc * icpg * ksz * ksz;
    float acc = 0.f;
    for (int ic = 0; ic < icpg; ++ic) {
        const float* ip = in + ((long)b * Cin + g * icpg + ic) * Nn;
        for (int ky = 0; ky < ksz; ++ky) {
            int yy = y + ky - pad; if (yy < 0 || yy >= 80) continue;
            for (int kx = 0; kx < ksz; ++kx) {
                int xx = x0 + kx - pad; if (xx < 0 || xx >= 80) continue;
                acc += ip[yy * 80 + xx] * wp[(ic * ksz + ky) * ksz + kx];
            }
        }
    }
    out[((long)b * outC + ocBase + oc) * Nn + n] = acc;
}

// Eval-mode BN (scale/shift) + ReLU on [B,R,N].
__global__ void etb_bnrelu(float* __restrict__ y, const float* __restrict__ g,
                           const float* __restrict__ bt, int R, long total)
{
    long i = (long)blockIdx.x * blockDim.x + threadIdx.x;
    if (i >= total) return;
    int ch = (int)((i / Nn) % R);
    float v = y[i] * (g[ch] * BNS) + bt[ch];
    y[i] = v > 0.f ? v : 0.f;
}

__global__ void etb_sigmoid_cmul(const float* __restrict__ z, float* __restrict__ re,
                                 float* __restrict__ im, long total)
{
    long i = (long)blockIdx.x * blockDim.x + threadIdx.x;
    if (i >= total) return;
    float s = 1.0f / (1.0f + expf(-z[i]));
    re[i] *= s; im[i] *= s;
}

__global__ void etb_abs(const float* __restrict__ re, const float* __restrict__ im,
                        float* __restrict__ out, float alpha, long total)
{
    long i = (long)blockIdx.x * blockDim.x + threadIdx.x;
    if (i >= total) return;
    out[i] = alpha * sqrtf(re[i] * re[i] + im[i] * im[i]);
}

// abs + scale + [k1][k2] -> index (k1 + 80*k2) scatter for the 6400-pt IFFT.
__global__ void etb_abs_tr(const float* __restrict__ re, const float* __restrict__ im,
                           float* __restrict__ out, float alpha, long total)
{
    long i = (long)blockIdx.x * blockDim.x + threadIdx.x;
    if (i >= total) return;
    long row = i / Nn; int j = (int)(i % Nn);
    int k1 = j / 80, k2 = j % 80;
    out[row * Nn + k1 + 80 * k2] = alpha * sqrtf(re[i] * re[i] + im[i] * im[i]);
}

__global__ void etb_twiddle(float* __restrict__ re, float* __restrict__ im,
                            const float* __restrict__ twr, const float* __restrict__ twi, long total)
{
    long i = (long)blockIdx.x * blockDim.x + threadIdx.x;
    if (i >= total) return;
    int j = (int)(i % Nn);
    float r = re[i], q = im[i], tr = twr[j], ti = twi[j];
    re[i] = r * tr - q * ti;
    im[i] = r * ti + q * tr;
}

// a = sigmoid(z)*|re+im*i| ; out = gelu_exact(a)*a
__global__ void etb_gate_abs_gelu(const float* __restrict__ z, const float* __restrict__ re,
                                  const float* __restrict__ im, float* __restrict__ out, long total)
{
    long i = (long)blockIdx.x * blockDim.x + threadIdx.x;
    if (i >= total) return;
    float s = 1.0f / (1.0f + expf(-z[i]));
    float a = s * sqrtf(re[i] * re[i] + im[i] * im[i]);
    out[i] = 0.5f * a * (1.0f + erff(a * 0.70710678118654752f)) * a;
}

__global__ void etb_gelu_mul(const float* __restrict__ x, float* __restrict__ out, long total)
{
    long i = (long)blockIdx.x * blockDim.x + threadIdx.x;
    if (i >= total) return;
    float a = x[i];
    out[i] = 0.5f * a * (1.0f + erff(a * 0.70710678118654752f)) * a;
}

__global__ void etb_add3(const float* __restrict__ a, const float* __restrict__ b,
                         const float* __restrict__ c, float* __restrict__ o, long total)
{
    long i = (long)blockIdx.x * blockDim.x + threadIdx.x;
    if (i >= total) return;
    o[i] = a[i] + b[i] + c[i];
}

// ---------------------------------------------------------------------------
// Host side
// ---------------------------------------------------------------------------
template<bool AIM, bool BIM, bool TRB, int NT, bool BOUND>
static inline void cg(hipStream_t st,
                      const float* Ar, const float* Ai, const float* Br, const float* Bi,
                      float* Cr, float* Ci, int M, int N, int K,
                      long sA, int lda, long sB, int ldb, long sC, int ldc,
                      int batch, const float* bias, int acc)
{
    dim3 grid((N + 16 * NT - 1) / (16 * NT), (M + 15) / 16, batch);
    cgemm16<AIM, BIM, TRB, NT, BOUND><<<grid, dim3(32), 0, st>>>(Ar, Ai, Br, Bi, Cr, Ci,
        M, N, K, sA, lda, sB, ldb, sC, ldc, bias, acc);
}

static inline dim3 NB(long n) { return dim3((unsigned)((n + 255) / 256)); }

extern "C" void kernel_launch(void* const* d_in, const int* in_sizes, int n_in,
                              void* d_out, int out_size, void* d_ws, size_t ws_size,
                              hipStream_t stream)
{
    (void)in_sizes; (void)n_in; (void)out_size; (void)ws_size;
    // ---- inputs (setup_inputs dict order) ----
    const float* X        = (const float*)d_in[0];
    const float* ln1_w    = (const float*)d_in[1];
    const float* ln2_w    = (const float*)d_in[2];
    const float* ln1_b    = (const float*)d_in[3];
    const float* ln2_b    = (const float*)d_in[4];
    const float* f_temp   = (const float*)d_in[5];
    const float* s_temp   = (const float*)d_in[6];
    const float* f_proj_w = (const float*)d_in[7];
    const float* s_proj_w = (const float*)d_in[8];
    const float* f_w1     = (const float*)d_in[9];
    const float* f_wb1    = (const float*)d_in[10];
    const float* f_bn_g   = (const float*)d_in[11];
    const float* f_bn_b   = (const float*)d_in[12];
    const float* f_w2     = (const float*)d_in[13];
    const float* f_wb2    = (const float*)d_in[14];
    const float* s_q1_w   = (const float*)d_in[15];
    const float* s_q1_b   = (const float*)d_in[16];
    const float* s_k1_w   = (const float*)d_in[17];
    const float* s_k1_b   = (const float*)d_in[18];
    const float* s_v1_w   = (const float*)d_in[19];
    const float* s_v1_b   = (const float*)d_in[20];
    const float* s_q3_w   = (const float*)d_in[21];
    const float* s_k3_w   = (const float*)d_in[22];
    const float* s_v3_w   = (const float*)d_in[23];
    const float* s_c3_w   = (const float*)d_in[24];
    const float* s_q5_w   = (const float*)d_in[25];
    const float* s_k5_w   = (const float*)d_in[26];
    const float* s_v5_w   = (const float*)d_in[27];
    const float* s_c5_w   = (const float*)d_in[28];
    const float* ff_dw1_w = (const float*)d_in[29];
    const float* ff_dw2_w = (const float*)d_in[30];
    const float* ff_proj_w= (const float*)d_in[31];
    const float* ffg_w1   = (const float*)d_in[32];
    const float* ffg_b1   = (const float*)d_in[33];
    const float* ffg_bng  = (const float*)d_in[34];
    const float* ffg_bnb  = (const float*)d_in[35];
    const float* ffg_w2   = (const float*)d_in[36];
    const float* ffg_b2   = (const float*)d_in[37];
    const float* ffg1_w1  = (const float*)d_in[38];
    const float* ffg1_b1  = (const float*)d_in[39];
    const float* ffg1_bng = (const float*)d_in[40];
    const float* ffg1_bnb = (const float*)d_in[41];
    const float* ffg1_w2  = (const float*)d_in[42];
    const float* ffg1_b2  = (const float*)d_in[43];

    // ---- workspace layout ----
    float* W = (float*)d_ws;
    float *F80r = W, *F80i = W + 6400, *iF80r = W + 12800, *iF80i = W + 19200;
    float *TWr = W + 25600, *TWi = W + 32000;
    float *iF32r = W + 38400, *iF32i = W + 39424;
    float *y8  = W + 40448;                 // [B,8,N] = 409600
    float *atR = W + 450048;                // [B,H,32,32] = 32768
    float *atI = W + 482816;
    float* big = W + 524288;
    auto S = [&](int i) { return big + (long)i * TOT; };

    etb_tables<<<25, 256, 0, stream>>>(F80r, F80i, iF80r, iF80i, TWr, TWi, iF32r, iF32i);

    // ======================= Attention phase =======================
    etb_ln<<<NB((long)Bn * Nn), 256, 0, stream>>>(X, ln1_w, ln1_b, S(0), (long)Bn * Nn);

    // xf = fft2(xn):  T = F80 * X ; xf = T * F80   (batch B*C = 1024 images)
    cg<true, false, false, 5, false>(stream, F80r, F80i, S(0), nullptr, S(1), S(2),
                           80, 80, 80, 0, 80, 6400, 80, 6400, 80, 1024, nullptr, 0);
    cg<true, true, false, 5, false>(stream, S(1), S(2), F80r, F80i, S(3), S(4),
                          80, 80, 80, 6400, 80, 0, 80, 6400, 80, 1024, nullptr, 0);

    // qf = l2norm(xf) rows of 6400 (complex)
    etb_l2norm<<<1024, 256, 0, stream>>>(S(3), S(4), S(5), S(6), Nn);

    // attn = qf . qf^T  (complex, per (b,h): 32x32, K=6400)
    cg<true, true, true, 2, false>(stream, S(5), S(6), S(5), S(6), atR, atI,
                         32, 32, 6400, CHROW, 6400, CHROW, 6400, 1024, 32, 32, nullptr, 0);
    etb_softmax32<<<1024, 32, 0, stream>>>(atR, f_temp, 1024);
    etb_softmax32<<<1024, 32, 0, stream>>>(atI, f_temp, 1024);

    // prod = attn . vf  (vf = xf)
    cg<true, true, false, 5, false>(stream, atR, atI, S(3), S(4), S(5), S(6),
                          32, 6400, 32, 1024, 32, CHROW, 6400, CHROW, 6400, 32, nullptr, 0);

    // ifft2 over (c=32, n=6400): step 1: 32-pt along c
    cg<true, true, false, 5, false>(stream, iF32r, iF32i, S(5), S(6), S(1), S(2),
                          32, 6400, 32, 0, 32, CHROW, 6400, CHROW, 6400, 32, nullptr, 0);
    // step 2: 6400-pt per row (Cooley-Tukey 80x80): Z = iF80*Y ; twiddle ; W = Z*iF80 ; scatter
    cg<true, true, false, 5, false>(stream, iF80r, iF80i, S(1), S(2), S(5), S(6),
                          80, 80, 80, 0, 80, 6400, 80, 6400, 80, 1024, nullptr, 0);
    etb_twiddle<<<NB(TOT), 256, 0, stream>>>(S(5), S(6), TWr, TWi, TOT);
    cg<true, true, false, 5, false>(stream, S(5), S(6), iF80r, iF80i, S(1), S(2),
                          80, 80, 80, 6400, 80, 0, 80, 6400, 80, 1024, nullptr, 0);
    etb_abs_tr<<<NB(TOT), 256, 0, stream>>>(S(1), S(2), S(7), 1.0f / (32.0f * 6400.0f), TOT); // out_f

    // gate on xf.real: conv1x1 -> BN+ReLU -> conv1x1 -> sigmoid * xf
    cg<false, false, false, 5, true>(stream, f_w1, nullptr, S(3), nullptr, y8, nullptr,
                            8, 6400, 128, 0, 128, PLANE, 6400, 8L * Nn, 6400, Bn, f_wb1, 0);
    etb_bnrelu<<<NB(8L * Bn * Nn), 256, 0, stream>>>(y8, f_bn_g, f_bn_b, 8, 8L * Bn * Nn);
    cg<false, false, false, 5, false>(stream, f_w2, nullptr, y8, nullptr, S(8), nullptr,
                            128, 6400, 8, 0, 8, 8L * Nn, 6400, PLANE, 6400, Bn, f_wb2, 0);
    etb_sigmoid_cmul<<<NB(TOT), 256, 0, stream>>>(S(8), S(3), S(4), TOT);
    // ifft2 over (h,w) + abs/6400 -> out_f_l
    cg<true, true, false, 5, false>(stream, iF80r, iF80i, S(3), S(4), S(5), S(6),
                          80, 80, 80, 0, 80, 6400, 80, 6400, 80, 1024, nullptr, 0);
    cg<true, true, false, 5, false>(stream, S(5), S(6), iF80r, iF80i, S(1), S(2),
                          80, 80, 80, 6400, 80, 0, 80, 6400, 80, 1024, nullptr, 0);
    etb_abs<<<NB(TOT), 256, 0, stream>>>(S(1), S(2), S(8), 1.0f / 6400.0f, TOT);

    // a_f = f_proj * [out_f ; out_f_l]  (K-split accumulate)
    cg<false, false, false, 5, false>(stream, f_proj_w, nullptr, S(7), nullptr, S(9), nullptr,
                            128, 6400, 128, 0, 256, PLANE, 6400, PLANE, 6400, Bn, nullptr, 0);
    cg<false, false, false, 5, false>(stream, f_proj_w + 128, nullptr, S(8), nullptr, S(9), nullptr,
                            128, 6400, 128, 0, 256, PLANE, 6400, PLANE, 6400, Bn, nullptr, 1);

    // ---- Attention_S ----
    long gc64 = (long)Bn * 64 * Nn;
    // q
    cg<false, false, false, 5, false>(stream, s_q1_w, nullptr, S(0), nullptr, S(10), nullptr,
                            128, 6400, 128, 0, 128, PLANE, 6400, PLANE, 6400, Bn, s_q1_b, 0);
    etb_gconv<<<NB(gc64), 256, 0, stream>>>(S(10), s_q3_w, S(11), 128, 64, 2, 1, 3, 1, 0, 128, gc64);
    etb_gconv<<<NB(gc64), 256, 0, stream>>>(S(10), s_q5_w, S(11), 128, 64, 2, 1, 5, 2, 64, 128, gc64);
    etb_l2norm<<<1024, 256, 0, stream>>>(S(11), nullptr, S(11), nullptr, Nn);
    // k
    cg<false, false, false, 5, false>(stream, s_k1_w, nullptr, S(0), nullptr, S(10), nullptr,
                            128, 6400, 128, 0, 128, PLANE, 6400, PLANE, 6400, Bn, s_k1_b, 0);
    etb_gconv<<<NB(gc64), 256, 0, stream>>>(S(10), s_k3_w, S(12), 128, 64, 2, 1, 3, 1, 0, 128, gc64);
    etb_gconv<<<NB(gc64), 256, 0, stream>>>(S(10), s_k5_w, S(12), 128, 64, 2, 1, 5, 2, 64, 128, gc64);
    etb_l2norm<<<1024, 256, 0, stream>>>(S(12), nullptr, S(12), nullptr, Nn);
    // attn_s
    cg<false, false, true, 2, false>(stream, S(11), nullptr, S(12), nullptr, atR, nullptr,
                           32, 32, 6400, CHROW, 6400, CHROW, 6400, 1024, 32, 32, nullptr, 0);
    etb_softmax32<<<1024, 32, 0, stream>>>(atR, s_temp, 1024);
    // v
    cg<false, false, false, 5, false>(stream, s_v1_w, nullptr, S(0), nullptr, S(10), nullptr,
                            128, 6400, 128, 0, 128, PLANE, 6400, PLANE, 6400, Bn, s_v1_b, 0);
    etb_gconv<<<NB(gc64), 256, 0, stream>>>(S(10), s_v3_w, S(13), 128, 64, 2, 1, 3, 1, 0, 128, gc64);
    etb_gconv<<<NB(gc64), 256, 0, stream>>>(S(10), s_v5_w, S(13), 128, 64, 2, 1, 5, 2, 64, 128, gc64);
    // out_s = attn_s . vs
    cg<false, false, false, 5, false>(stream, atR, nullptr, S(13), nullptr, S(10), nullptr,
                            32, 6400, 32, 1024, 32, CHROW, 6400, CHROW, 6400, 32, nullptr, 0);
    // out_s_l
    etb_gconv<<<NB(gc64), 256, 0, stream>>>(S(0), s_c3_w, S(12), 128, 64, 2, 1, 3, 1, 0, 128, gc64);
    etb_gconv<<<NB(gc64), 256, 0, stream>>>(S(0), s_c5_w, S(12), 128, 64, 2, 1, 5, 2, 64, 128, gc64);
    // a_s
    cg<false, false, false, 5, false>(stream, s_proj_w, nullptr, S(10), nullptr, S(11), nullptr,
                            128, 6400, 128, 0, 256, PLANE, 6400, PLANE, 6400, Bn, nullptr, 0);
    cg<false, false, false, 5, false>(stream, s_proj_w + 128, nullptr, S(12), nullptr, S(11), nullptr,
                            128, 6400, 128, 0, 256, PLANE, 6400, PLANE, 6400, Bn, nullptr, 1);

    // x1 = x + a_f + a_s
    etb_add3<<<NB(TOT), 256, 0, stream>>>(X, S(9), S(11), S(14), TOT);

    // ======================= FeedForward phase =======================
    etb_ln<<<NB((long)Bn * Nn), 256, 0, stream>>>(S(14), ln2_w, ln2_b, S(0), (long)Bn * Nn);

    // xf2 = fft2(xn2)
    cg<true, false, false, 5, false>(stream, F80r, F80i, S(0), nullptr, S(1), S(2),
                           80, 80, 80, 0, 80, 6400, 80, 6400, 80, 1024, nullptr, 0);
    cg<true, true, false, 5, false>(stream, S(1), S(2), F80r, F80i, S(3), S(4),
                          80, 80, 80, 6400, 80, 0, 80, 6400, 80, 1024, nullptr, 0);
    // ffg gate -> z
    cg<false, false, false, 5, true>(stream, ffg_w1, nullptr, S(3), nullptr, y8, nullptr,
                            8, 6400, 128, 0, 128, PLANE, 6400, 8L * Nn, 6400, Bn, ffg_b1, 0);
    etb_bnrelu<<<NB(8L * Bn * Nn), 256, 0, stream>>>(y8, ffg_bng, ffg_bnb, 8, 8L * Bn * Nn);
    cg<false, false, false, 5, false>(stream, ffg_w2, nullptr, y8, nullptr, S(5), nullptr,
                            128, 6400, 8, 0, 8, 8L * Nn, 6400, PLANE, 6400, Bn, ffg_b2, 0);
    // cat lower half: x_f_g = gelu(s*|xf2|) * (s*|xf2|)
    etb_gate_abs_gelu<<<NB(TOT), 256, 0, stream>>>(S(5), S(3), S(4), S(6), TOT);
    // cat upper half: x_s_g = gelu(dwconv3(xn2)) * dwconv3(xn2)
    etb_gconv<<<NB(TOT), 256, 0, stream>>>(S(0), ff_dw1_w, S(1), 128, 128, 1, 1, 3, 1, 0, 128, TOT);
    etb_gelu_mul<<<NB(TOT), 256, 0, stream>>>(S(1), S(7), TOT);

    // cf = fft2(cat)  (batch B*256 = 2048 images; cat contiguous at S6..S7)
    cg<true, false, false, 5, false>(stream, F80r, F80i, S(6), nullptr, S(0), S(2),
                           80, 80, 80, 0, 80, 6400, 80, 6400, 80, 2048, nullptr, 0);
    cg<true, true, false, 5, false>(stream, S(0), S(2), F80r, F80i, S(8), S(10),
                          80, 80, 80, 6400, 80, 0, 80, 6400, 80, 2048, nullptr, 0);

    // ffg1 gate on cf.real (K=256 split into 2x128) -> z3 -> sigmoid * cf
    cg<false, false, false, 5, true>(stream, ffg1_w1, nullptr, S(8), nullptr, y8, nullptr,
                            8, 6400, 128, 0, 256, 2 * PLANE, 6400, 8L * Nn, 6400, Bn, ffg1_b1, 0);
    cg<false, false, false, 5, true>(stream, ffg1_w1 + 128, nullptr, S(8) + 128L * Nn, nullptr, y8, nullptr,
                            8, 6400, 128, 0, 256, 2 * PLANE, 6400, 8L * Nn, 6400, Bn, nullptr, 1);
    etb_bnrelu<<<NB(8L * Bn * Nn), 256, 0, stream>>>(y8, ffg1_bng, ffg1_bnb, 8, 8L * Bn * Nn);
    cg<false, false, false, 5, false>(stream, ffg1_w2, nullptr, y8, nullptr, S(4), nullptr,
                            256, 6400, 8, 0, 8, 8L * Nn, 6400, 2 * PLANE, 6400, Bn, ffg1_b2, 0);
    etb_sigmoid_cmul<<<NB(2 * TOT), 256, 0, stream>>>(S(4), S(8), S(10), 2 * TOT);

    // x_f2b = abs(ifft2(...)) / 6400
    cg<true, true, false, 5, false>(stream, iF80r, iF80i, S(8), S(10), S(0), S(2),
                          80, 80, 80, 0, 80, 6400, 80, 6400, 80, 2048, nullptr, 0);
    cg<true, true, false, 5, false>(stream, S(0), S(2), iF80r, iF80i, S(12), S(4),
                          80, 80, 80, 6400, 80, 0, 80, 6400, 80, 2048, nullptr, 0);
    etb_abs<<<NB(2 * TOT), 256, 0, stream>>>(S(12), S(4), S(8), 1.0f / 6400.0f, 2 * TOT);

    // x_s2b = grouped dwconv3 on cat (groups=128, 2 in / 2 out per group)
    etb_gconv<<<NB(2 * TOT), 256, 0, stream>>>(S(6), ff_dw2_w, S(10), 256, 256, 2, 2, 3, 1, 0, 256, 2 * TOT);

    // out = x1 + ff_proj * [x_f2b ; x_s2b]
    hipMemcpyAsync(d_out, S(14), TOT * sizeof(float), hipMemcpyDeviceToDevice, stream);
    cg<false, false, false, 5, false>(stream, ff_proj_w, nullptr, S(8), nullptr, (float*)d_out, nullptr,
                            128, 6400, 256, 0, 512, 2 * PLANE, 6400, PLANE, 6400, Bn, nullptr, 1);
    cg<false, false, false, 5, false>(stream, ff_proj_w + 256, nullptr, S(10), nullptr, (float*)d_out, nullptr,
                            128, 6400, 256, 0, 512, 2 * PLANE, 6400, PLANE, 6400, Bn, nullptr, 1);
}